// RoiPoolingConv_70368744178168
// MI455X (gfx1250) — compile-verified
//
#include <hip/hip_runtime.h>
#include <hip/hip_bf16.h>

// ROI-Align (bilinear, 7x7 pool) expressed as fp32 WMMA GEMM:
//   out(49 x C) = K(49 x 196) x G(196 x C)  per (roi, batch)
// K = sparse bilinear weight matrix (4 nz/row), G = gathered corner rows.
// 196 = 49 * 4 -> exactly 49 chained V_WMMA_F32_16X16X4_F32 steps.
// G gather uses GLOBAL_LOAD_ASYNC_TO_LDS_B128 (ASYNCcnt-tracked, GVS mode).

typedef __attribute__((ext_vector_type(2))) float v2f;
typedef __attribute__((ext_vector_type(8))) float v8f;

#define POOLP 7
#define AXMAX 14          // max distinct corner coords per axis (2 per sample x 7)
#define SDIM  196         // AXMAX * AXMAX, == 49 * 4
#define KSTEPS 49         // SDIM / 4
#define SK 197            // K_lds row stride (floats) -> spreads A-frag banks
#define SG 80             // G_lds row stride (floats), 16B-aligned rows
#define NC 64             // channels per block
#define KFLOATS (64 * SK)         // 64 padded M rows
#define GFLOATS (SDIM * SG)

__device__ __forceinline__ void setup_axis(int start, int size,
                                           int* g, int* i0, int* i1, float* w) {
    int n = 0;
    for (int p = 0; p < POOLP; ++p) {
        float coord = (p + 0.5f) * ((float)size / 7.0f) - 0.5f;
        float f = floorf(coord);
        int lo = (int)f; if (lo < 0) lo = 0;
        int hi = (int)ceilf(coord);
        if (hi < 0) hi = 0;
        if (hi > size - 1) hi = size - 1;
        w[p] = coord - f;
        int a0 = start + lo, a1 = start + hi;
        int k0 = -1, k1 = -1;
        for (int q = 0; q < n; ++q) if (g[q] == a0) k0 = q;
        if (k0 < 0) { g[n] = a0; k0 = n++; }
        for (int q = 0; q < n; ++q) if (g[q] == a1) k1 = q;
        if (k1 < 0) { g[n] = a1; k1 = n++; }
        i0[p] = k0; i1[p] = k1;
    }
    for (int q = n; q < AXMAX; ++q) g[q] = g[n - 1];   // pad -> in-bounds loads
}

__global__ __launch_bounds__(256)
void roi_align_wmma_kernel(const float* __restrict__ img,
                           const int* __restrict__ rois,
                           float* __restrict__ out) {
    extern __shared__ char smem[];
    float* Kl  = (float*)smem;                               // 64 x SK
    float* Gl  = (float*)(smem + KFLOATS * 4);               // SDIM x SG
    int*   gyx = (int*)(smem + (KFLOATS + GFLOATS) * 4);     // gy[14], gx[14]
    int*   idx = gyx + 2 * AXMAX;                            // iy0[7] iy1[7] ix0[7] ix1[7]
    float* wgt = (float*)(idx + 4 * POOLP);                  // wy[7], wx[7]

    const int tid = threadIdx.x;
    const int rb  = blockIdx.x;           // rb = r*8 + b  (matches output layout)
    const int r   = rb >> 3;
    const int b   = rb & 7;
    const int c0  = blockIdx.y * NC;

    // ---- stage 0: axis setup (threads 0,1) + zero K (all threads) ----
    if (tid == 0) {
        int rx = rois[r * 4 + 0], rw = rois[r * 4 + 2];
        setup_axis(rx, rw, gyx + AXMAX, idx + 2 * POOLP, idx + 3 * POOLP, wgt + POOLP); // x
    } else if (tid == 1) {
        int ry = rois[r * 4 + 1], rh = rois[r * 4 + 3];
        setup_axis(ry, rh, gyx, idx, idx + POOLP, wgt);                                  // y
    }
    for (int j = tid; j < KFLOATS; j += 256) Kl[j] = 0.0f;
    __syncthreads();

    // ---- stage 1: fill K (49 threads) + async-gather G rows into LDS ----
    if (tid < 49) {
        int py = tid / 7, px = tid % 7;
        float wy = wgt[py], wx = wgt[POOLP + px];
        int y0 = idx[py],             y1 = idx[POOLP + py];
        int x0 = idx[2 * POOLP + px], x1 = idx[3 * POOLP + px];
        float* row = Kl + tid * SK;
        row[y0 * AXMAX + x0] += (1.0f - wy) * (1.0f - wx);
        row[y0 * AXMAX + x1] += (1.0f - wy) * wx;
        row[y1 * AXMAX + x0] += wy * (1.0f - wx);
        row[y1 * AXMAX + x1] += wy * wx;
    }
    // G: SDIM rows x NC floats, 16B per lane via async global->LDS copies.
    // Dynamic LDS begins right after static LDS (== groupstaticsize).
    {
        const unsigned gl_base = __builtin_amdgcn_groupstaticsize()
                               + (unsigned)(KFLOATS * 4);
        for (int k = 0; k < (SDIM * (NC / 4) + 255) / 256; ++k) {
            int chunk = tid + k * 256;
            if (chunk < SDIM * (NC / 4)) {
                int row = chunk >> 4;           // NC/4 == 16 chunks per row
                int c4  = chunk & 15;
                int yi = row / AXMAX, xi = row % AXMAX;
                unsigned goff = ((unsigned)((b * 64 + gyx[yi]) * 64 + gyx[AXMAX + xi])
                                 * 1024u + (unsigned)(c0 + c4 * 4)) * 4u;   // bytes
                unsigned ldsb = gl_base + (unsigned)(row * SG + c4 * 4) * 4u;
                asm volatile("global_load_async_to_lds_b128 %0, %1, %2"
                             :: "v"(ldsb), "v"(goff), "s"(img)
                             : "memory");
            }
        }
        asm volatile("s_wait_asynccnt 0x0" ::: "memory");
    }
    __syncthreads();

    // ---- stage 2: WMMA chain, one wave per (M-tile, N-tile pair) ----
    const int wv   = tid >> 5;            // 0..7 (wave32)
    const int lane = tid & 31;
    const int half = lane >> 4;
    const int li   = lane & 15;
    const int mt   = wv & 3;              // M-tile 0..3 (rows 16*mt .. 16*mt+15)

    for (int i = 0; i < 2; ++i) {
        int nt = (wv >> 2) * 2 + i;       // N-tile 0..3 (16 channels each)
        v8f acc = {};
        const float* arow = Kl + (mt * 16 + li) * SK;
        const float* bcol = Gl + nt * 16 + li;
        for (int t = 0; t < KSTEPS; ++t) {
            int kb = t * 4 + half * 2;
            v2f a, bf;
            a.x  = arow[kb];
            a.y  = arow[kb + 1];
            bf.x = bcol[kb * SG];
            bf.y = bcol[(kb + 1) * SG];
            acc = __builtin_amdgcn_wmma_f32_16x16x4_f32(
                false, a, false, bf, (short)0, acc, false, false);
        }
        // store D: lane<16 -> M = mt*16 + v ; lane>=16 -> M = mt*16 + v + 8.
        // M-tiles 0..2: all 16 rows valid (wave-uniform, no exec churn).
        // M-tile 3: only row 48 valid (v==0, half==0).
        int c = c0 + nt * 16 + li;
        float* obase = out + (size_t)rb * 49 * 1024 + c;
        if (mt < 3) {
#pragma unroll
            for (int v = 0; v < 8; ++v) {
                obase[(size_t)(mt * 16 + v + 8 * half) * 1024] = acc[v];
            }
        } else if (half == 0) {
            obase[(size_t)48 * 1024] = acc[0];
        }
    }
}

extern "C" void kernel_launch(void* const* d_in, const int* in_sizes, int n_in,
                              void* d_out, int out_size, void* d_ws, size_t ws_size,
                              hipStream_t stream) {
    const float* img  = (const float*)d_in[0];   // (8, 64, 64, 1024) fp32
    const int*   rois = (const int*)d_in[1];     // (32, 4) int32
    float*       out  = (float*)d_out;           // (32, 8, 7, 7, 1024) fp32

    const size_t shmem = (size_t)(KFLOATS + GFLOATS) * 4      // K + G
                         + (2 * AXMAX + 4 * POOLP) * 4        // gy/gx + idx
                         + 2 * POOLP * 4;                     // weights
    dim3 grid(256, 1024 / NC);   // 256 (roi,batch) pairs x 16 channel slices
    roi_align_wmma_kernel<<<grid, 256, shmem, stream>>>(img, rois, out);
}